// approx_Conv2d_int8_5239860101335
// MI455X (gfx1250) — compile-verified
//
#include <hip/hip_runtime.h>
#include <stdint.h>

// Problem constants (match reference)
static constexpr int CIN  = 64;
static constexpr int COUT = 128;
static constexpr int Kdim = 576;    // CIN*3*3
static constexpr int Mdim = 4096;   // B*OH*OW

__device__ __forceinline__ int quant_idx(float v, float scale) {
    // matches jnp.clip(jnp.round(v/scale), -128, 127) + 128  (RNE rounding)
    float r = rintf(v / scale);
    r = fminf(127.0f, fmaxf(-128.0f, r));
    return (int)r + 128;
}

// ---------------------------------------------------------------------------
// Kernel 1: quantize im2col activations -> qa[m, k]  (uint8 LUT indices)
// grid: 4096 blocks (m), 576 threads (k).  Zero padding -> index 128.
// ---------------------------------------------------------------------------
__global__ void __launch_bounds__(576)
k_quant_a(const float* __restrict__ x, uint8_t* __restrict__ qa) {
    int k = threadIdx.x;            // 0..575  (c*9 + kh*3 + kw)
    int m = blockIdx.x;             // 0..4095 ((b*32+oh)*32+ow)
    int c  = k / 9;
    int r  = k - c * 9;
    int kh = r / 3;
    int kw = r - kh * 3;
    int ow = m & 31, oh = (m >> 5) & 31, b = m >> 10;
    int ih = oh + kh - 1, iw = ow + kw - 1;
    float v = 0.0f;
    if ((unsigned)ih < 32u && (unsigned)iw < 32u)
        v = x[(((b * CIN + c) * 32 + ih) * 32) + iw];
    qa[(size_t)m * Kdim + k] = (uint8_t)quant_idx(v, 0.05f);
}

// ---------------------------------------------------------------------------
// Kernel 2: quantize weights -> qb[k, n]
// ---------------------------------------------------------------------------
__global__ void __launch_bounds__(256)
k_quant_b(const float* __restrict__ w, uint8_t* __restrict__ qb) {
    int idx = blockIdx.x * blockDim.x + threadIdx.x;   // k*128 + n
    if (idx >= Kdim * COUT) return;
    int n = idx & 127;
    int k = idx >> 7;
    qb[idx] = (uint8_t)quant_idx(w[n * Kdim + k], 0.03f);
}

// ---------------------------------------------------------------------------
// Kernel 3: build gathered table  T[k, v, n] = lut[v, qb[k,n]]
// 75.5 MB -> stays L2-resident (192 MB L2).  Coalesced 512B stores.
// grid: 576 blocks (k), 128 threads (n)
// ---------------------------------------------------------------------------
__global__ void __launch_bounds__(128)
k_build_T(const uint8_t* __restrict__ qb,
          const float*  __restrict__ lut,
          float*        __restrict__ T) {
    int k = blockIdx.x;
    int n = threadIdx.x;
    int qbv = qb[k * COUT + n];
    const float* lcol = lut + qbv;                    // column qbv of lut
    float* trow = T + (size_t)k * 256 * COUT + n;
    #pragma unroll 4
    for (int v = 0; v < 256; ++v)
        trow[v * COUT] = lcol[v * 256];               // lut[v*256 + qbv]
}

// ---------------------------------------------------------------------------
// Kernel 4 (tier 1): main accumulation via T-table.
// One wave per m; lane l owns n = 4l..4l+3 (one coalesced float4 of each T row).
// qa row is wave-uniform; uint4 chunks -> s_load_b128 (16 indices / scalar load,
// one kmcnt wait per 16 k's).  bias is DMA'd into LDS once per block via
// global_load_async_to_lds_b128 (CDNA5 async path).
// grid: 512 blocks x 256 threads (8 waves)
// ---------------------------------------------------------------------------
__global__ void __launch_bounds__(256)
k_main_T(const uint8_t* __restrict__ qa,
         const float*  __restrict__ T,
         const float*  __restrict__ bias,
         float*        __restrict__ out) {
    __shared__ float bias_lds[COUT];
    const int lane = threadIdx.x & 31;
    const int wv   = threadIdx.x >> 5;

    // --- async DMA: 32 lanes x 16B = 512B of bias -> LDS ---
    if (threadIdx.x < 32) {
        uint32_t lo = (uint32_t)(uintptr_t)(&bias_lds[lane * 4]); // LDS byte offset (addr[31:0])
        uint64_t ga = (uint64_t)(uintptr_t)(bias + lane * 4);
        asm volatile("global_load_async_to_lds_b128 %0, %1, off"
                     :: "v"(lo), "v"(ga) : "memory");
    }
    asm volatile("s_wait_asynccnt 0x0" ::: "memory");
    __syncthreads();

    const int m = __builtin_amdgcn_readfirstlane(blockIdx.x * 8 + wv);
    const uint4* qa128 = (const uint4*)(qa + (size_t)m * Kdim); // SGPR base -> s_load_b128
    const char* Tb = (const char*)T;
    const uint32_t loff = (uint32_t)lane * 16u;

    float4 acc0 = make_float4(0.f, 0.f, 0.f, 0.f);
    float4 acc1 = make_float4(0.f, 0.f, 0.f, 0.f);
    for (int c = 0; c < Kdim / 16; ++c) {            // 36 chunks of 16 k's
        uint4 a16 = qa128[c];                        // 16 LUT indices, one s_load_b128
        uint32_t cb = (uint32_t)c * 2097152u;        // 16 * 256 * 128 * 4 bytes
        uint32_t dv[4] = {a16.x, a16.y, a16.z, a16.w};
        #pragma unroll
        for (int d = 0; d < 4; ++d) {                // 4 k's per dword
            uint32_t a4 = dv[d];
            uint32_t kb = cb + (uint32_t)d * 524288u;
            #pragma unroll
            for (int j = 0; j < 4; ++j) {
                uint32_t a = (a4 >> (8 * j)) & 255u;
                const float4* p = (const float4*)(Tb + (kb + (uint32_t)j * 131072u
                                                          + a * 512u + loff));
                float4 t = *p;                       // coalesced 512B/wave
                if (j & 1) { acc1.x += t.x; acc1.y += t.y; acc1.z += t.z; acc1.w += t.w; }
                else       { acc0.x += t.x; acc0.y += t.y; acc0.z += t.z; acc0.w += t.w; }
            }
        }
    }

    const float osc = (float)(0.05 * 0.03);
    int b = m >> 10, rem = m & 1023;
    float* op = out + (size_t)b * COUT * 1024 + rem;    // out[b][n][oh][ow]
    int n0 = lane * 4;
    op[(size_t)(n0 + 0) * 1024] = (acc0.x + acc1.x) * osc + bias_lds[n0 + 0];
    op[(size_t)(n0 + 1) * 1024] = (acc0.y + acc1.y) * osc + bias_lds[n0 + 1];
    op[(size_t)(n0 + 2) * 1024] = (acc0.z + acc1.z) * osc + bias_lds[n0 + 2];
    op[(size_t)(n0 + 3) * 1024] = (acc0.w + acc1.w) * osc + bias_lds[n0 + 3];
}

// ---------------------------------------------------------------------------
// Tier 2: qa/qb precomputed but no room for T -> gather lut (L2) directly.
// ---------------------------------------------------------------------------
__global__ void __launch_bounds__(256)
k_main_direct(const uint8_t* __restrict__ qa,
              const uint8_t* __restrict__ qb,
              const float*  __restrict__ lut,
              const float*  __restrict__ bias,
              float*        __restrict__ out) {
    const int lane = threadIdx.x & 31;
    const int m = __builtin_amdgcn_readfirstlane(blockIdx.x * 8 + (threadIdx.x >> 5));
    const uint4* qa128 = (const uint4*)(qa + (size_t)m * Kdim);
    float4 acc = make_float4(0.f, 0.f, 0.f, 0.f);
    for (int c = 0; c < Kdim / 16; ++c) {
        uint4 a16 = qa128[c];
        uint32_t dv[4] = {a16.x, a16.y, a16.z, a16.w};
        #pragma unroll
        for (int d = 0; d < 4; ++d) {
            uint32_t a4 = dv[d];
            #pragma unroll
            for (int j = 0; j < 4; ++j) {
                int k = c * 16 + d * 4 + j;
                uint32_t a = (a4 >> (8 * j)) & 255u;
                const float* lrow = lut + a * 256;
                uchar4 b4 = *(const uchar4*)(qb + k * COUT + lane * 4);
                acc.x += lrow[b4.x]; acc.y += lrow[b4.y];
                acc.z += lrow[b4.z]; acc.w += lrow[b4.w];
            }
        }
    }
    const float osc = (float)(0.05 * 0.03);
    int b = m >> 10, rem = m & 1023;
    float* op = out + (size_t)b * COUT * 1024 + rem;
    int n0 = lane * 4;
    op[(size_t)(n0 + 0) * 1024] = acc.x * osc + bias[n0 + 0];
    op[(size_t)(n0 + 1) * 1024] = acc.y * osc + bias[n0 + 1];
    op[(size_t)(n0 + 2) * 1024] = acc.z * osc + bias[n0 + 2];
    op[(size_t)(n0 + 3) * 1024] = acc.w * osc + bias[n0 + 3];
}

// ---------------------------------------------------------------------------
// Tier 3: no workspace at all (safety net). One thread per (m, n).
// ---------------------------------------------------------------------------
__global__ void __launch_bounds__(256)
k_main_nows(const float* __restrict__ x, const float* __restrict__ w,
            const float* __restrict__ lut, const float* __restrict__ bias,
            float* __restrict__ out) {
    int idx = blockIdx.x * blockDim.x + threadIdx.x;
    if (idx >= Mdim * COUT) return;
    int n = idx & 127, m = idx >> 7;
    int ow = m & 31, oh = (m >> 5) & 31, b = m >> 10;
    float acc = 0.f;
    for (int c = 0; c < CIN; ++c)
        for (int kh = 0; kh < 3; ++kh)
            for (int kw = 0; kw < 3; ++kw) {
                int ih = oh + kh - 1, iw = ow + kw - 1;
                float xv = ((unsigned)ih < 32u && (unsigned)iw < 32u)
                               ? x[(((b * CIN + c) * 32 + ih) * 32) + iw] : 0.f;
                int a  = quant_idx(xv, 0.05f);
                int bb = quant_idx(w[((n * CIN + c) * 3 + kh) * 3 + kw], 0.03f);
                acc += lut[a * 256 + bb];
            }
    out[((size_t)b * COUT + n) * 1024 + (m & 1023)] = acc * (float)(0.05 * 0.03) + bias[n];
}

// ---------------------------------------------------------------------------
extern "C" void kernel_launch(void* const* d_in, const int* in_sizes, int n_in,
                              void* d_out, int out_size, void* d_ws, size_t ws_size,
                              hipStream_t stream) {
    const float* x    = (const float*)d_in[0];
    const float* w    = (const float*)d_in[1];
    const float* bias = (const float*)d_in[2];
    const float* lut  = (const float*)d_in[3];
    // d_in[4] = gradient_lut (unused in forward)
    float* out = (float*)d_out;

    const size_t QA_BYTES = (size_t)Mdim * Kdim;           // 2,359,296
    const size_t QB_BYTES = (size_t)Kdim * COUT;           //    73,728
    const size_t T_BYTES  = (size_t)Kdim * 256 * COUT * 4; // 75,497,472
    const size_t NEED_SMALL = QA_BYTES + QB_BYTES;
    const size_t NEED_T     = NEED_SMALL + T_BYTES;

    uint8_t* qa = (uint8_t*)d_ws;
    uint8_t* qb = qa + QA_BYTES;
    float*   T  = (float*)(qb + QB_BYTES);                 // 16B-aligned offset

    if (ws_size >= NEED_T) {
        k_quant_a<<<Mdim, Kdim, 0, stream>>>(x, qa);
        k_quant_b<<<(Kdim * COUT + 255) / 256, 256, 0, stream>>>(w, qb);
        k_build_T<<<Kdim, COUT, 0, stream>>>(qb, lut, T);
        k_main_T<<<Mdim / 8, 256, 0, stream>>>(qa, T, bias, out);
    } else if (ws_size >= NEED_SMALL) {
        k_quant_a<<<Mdim, Kdim, 0, stream>>>(x, qa);
        k_quant_b<<<(Kdim * COUT + 255) / 256, 256, 0, stream>>>(w, qb);
        k_main_direct<<<Mdim / 8, 256, 0, stream>>>(qa, qb, lut, bias, out);
    } else {
        k_main_nows<<<(Mdim * COUT + 255) / 256, 256, 0, stream>>>(x, w, lut, bias, out);
    }
}